// NMS_7069516169423
// MI455X (gfx1250) — compile-verified
//
#include <hip/hip_runtime.h>
#include <hip/hip_bf16.h>
#include <stdint.h>

// ---------------------------------------------------------------------------
// NMS for (32, 25200, 16) predictions on MI455X (gfx1250, wave32).
// One workgroup (1024 threads = 32 wave32 waves) per image.
//   A: stream scores, pack u64 sort keys into LDS (256KB fits CDNA5 320KB LDS)
//   B: bitonic TOP-K: sort 32 chunks of 1024 descending, then 5 rounds of
//      Batcher k-selection merge (A[i] = max(A[i], B[1023-i]) + bitonic
//      clean-up) -> top-1024 sorted descending at LDS offset 0.
//   C: gather top-1024 boxes, xyxy + area into LDS (float4 rows -> b128)
//   D: 16x16 IoU tiles (upper triangle only); area_i+area_j via
//      V_WMMA_F32_16X16X4_F32 outer-product; division-free compare;
//      boxes fetched as ds_load_b128; masks assembled via wave32 ballot
//   E: serial greedy suppression scan on one wave (1024-bit keep in 32 lanes)
//   F: emit det rows with b128 gathers/stores, plus keep flags
// ---------------------------------------------------------------------------

#define NPRED  25200
#define NSORT  32768
#define TOPK   1024
#define NTILE  64            // 1024 / 16
#define CONF_T 0.25f
#define IOU_T  0.45f

typedef float v2f __attribute__((ext_vector_type(2)));
typedef float v8f __attribute__((ext_vector_type(8)));

// LDS layout (bytes), dynamic extern shared of 262144:
//   [0,      262144) u64 keys[32768]                (phases A,B)
//   -- after top-k, overlaid: --
//   [0,       16384) float4 boxes[1024]
//   [16384,   20480) float  areas[1024]
//   [20480,   24576) float  scores[1024]
//   [24576,   28672) uint   idxs[1024]
//   [28672,  159744) u16    mask16[1024][64]
//   [159744, 159872) uint   keepW[32]
#define OFF_AREAS  16384
#define OFF_SCORES 20480
#define OFF_IDXS   24576
#define OFF_MASK   28672
#define OFF_KEEPW  (28672 + 131072)

__global__ __launch_bounds__(1024, 1)
void nms_gfx1250_kernel(const float* __restrict__ pred,
                        float* __restrict__ outDet,
                        float* __restrict__ outKeep)
{
    extern __shared__ unsigned char smem[];
    unsigned long long* keys   = reinterpret_cast<unsigned long long*>(smem);
    float4*             boxes4 = reinterpret_cast<float4*>(smem);
    float*              areas  = reinterpret_cast<float*>(smem + OFF_AREAS);
    float*              scores = reinterpret_cast<float*>(smem + OFF_SCORES);
    unsigned*           idxs   = reinterpret_cast<unsigned*>(smem + OFF_IDXS);
    unsigned short*     mask16 = reinterpret_cast<unsigned short*>(smem + OFF_MASK);
    unsigned*           keepW  = reinterpret_cast<unsigned*>(smem + OFF_KEEPW);

    const int b   = blockIdx.x;
    const int tid = threadIdx.x;
    const float* P = pred + (size_t)b * NPRED * 16;

    // ---------------- Phase A: build sort keys ----------------
    // key = (monotone-ordered score bits) << 32 | (0xFFFFFFFF - idx)
    // Descending order => highest score first, ties by lower index,
    // matching jax.lax.top_k. Padding entries get key 0 (below any real key).
    for (int n = tid; n < NSORT; n += 1024) {
        unsigned long long key = 0ull;
        if (n < NPRED) {
            const float* row = P + (size_t)n * 16;
            __builtin_prefetch(row + 1024 * 16, 0, 1);   // global_prefetch_b8
            float obj = row[4];
            float cls = row[15];
            float sc  = obj * cls;
            bool valid = (obj > CONF_T) && (sc > CONF_T);
            float s = valid ? sc : -1.0f;
            unsigned ub  = __float_as_uint(s);
            unsigned ord = ub ^ ((ub >> 31) ? 0xFFFFFFFFu : 0x80000000u);
            key = ((unsigned long long)ord << 32) |
                  (unsigned long long)(0xFFFFFFFFu - (unsigned)n);
        }
        keys[n] = key;
    }
    __syncthreads();

    // ---------------- Phase B1: sort each 1024-chunk descending -------------
    // For k < 1024, i&k equals chunk-local il&k; for k == 1024, (k & 1023)==0
    // forces descending in every chunk.
    for (unsigned k = 2; k <= 1024; k <<= 1) {
        for (unsigned j = k >> 1; j > 0; j >>= 1) {
            for (unsigned v = tid; v < NSORT / 2; v += 1024) {
                unsigned i = 2u * (v & ~(j - 1u)) + (v & (j - 1u));
                unsigned p = i | j;
                unsigned long long a = keys[i];
                unsigned long long c = keys[p];
                bool desc = ((i & k & 1023u) == 0u);
                if (desc ? (a < c) : (a > c)) { keys[i] = c; keys[p] = a; }
            }
            __syncthreads();
        }
    }

    // ---------------- Phase B2: 5 rounds of k-selection merge ---------------
    for (int r = 0; r < 5; ++r) {
        unsigned nmerge = 16u >> r;
        unsigned span   = 2048u << r;
        unsigned half   = 1024u << r;

        for (unsigned v = tid; v < nmerge * 1024u; v += 1024) {
            unsigned m = v >> 10;
            unsigned e = v & 1023u;
            unsigned ai = m * span + e;
            unsigned bi = m * span + half + (1023u - e);
            unsigned long long a = keys[ai];
            unsigned long long c = keys[bi];
            keys[ai] = (a > c) ? a : c;
        }
        __syncthreads();

        for (unsigned j = 512; j >= 1; j >>= 1) {
            for (unsigned v = tid; v < nmerge * 512u; v += 1024) {
                unsigned m  = v >> 9;
                unsigned p  = v & 511u;
                unsigned il = 2u * (p & ~(j - 1u)) + (p & (j - 1u));
                unsigned i  = m * span + il;
                unsigned q  = i | j;
                unsigned long long a = keys[i];
                unsigned long long c = keys[q];
                if (a < c) { keys[i] = c; keys[q] = a; }   // descending
            }
            __syncthreads();
        }
    }
    // Top-1024, sorted descending, now lives at keys[0..1023].

    // ---------------- Phase C: decode top-1024, gather boxes ----------------
    unsigned long long myKey = keys[tid];
    __syncthreads();      // keys region is overlaid below from here on
    {
        unsigned idx = 0xFFFFFFFFu - (unsigned)(myKey & 0xFFFFFFFFull);
        unsigned ord = (unsigned)(myKey >> 32);
        unsigned ub  = ord ^ ((ord >> 31) ? 0x80000000u : 0xFFFFFFFFu);
        float s = __uint_as_float(ub);
        float cx = 0.f, cy = 0.f, w = 0.f, h = 0.f;
        if (idx < NPRED) {
            const float4 q0 = *reinterpret_cast<const float4*>(P + (size_t)idx * 16);
            cx = q0.x; cy = q0.y; w = q0.z; h = q0.w;
        } else {
            idx = 0;
        }
        float x1 = cx - 0.5f * w, y1 = cy - 0.5f * h;
        float x2 = cx + 0.5f * w, y2 = cy + 0.5f * h;
        boxes4[tid] = make_float4(x1, y1, x2, y2);
        areas[tid]  = (x2 - x1) * (y2 - y1);
        scores[tid] = s;
        idxs[tid]   = idx;
    }
    __syncthreads();

    // ---------------- Phase D: IoU tiles with WMMA area-sum ----------------
    // Tile (I,J): rows I*16..+15, cols J*16..+15. One wave per tile.
    // S = A(16x4) x B(4x16): A row m = [area_i(m), 1, 0, 0],
    //                        B col n = [1, area_j(n), 0, 0]^T
    // => S[m][n] = area_i(m) + area_j(n)  (V_WMMA_F32_16X16X4_F32).
    // C/D layout: VGPR r, lane l -> row = r + 8*(l>=16), col = l%16.
    // Lower-triangle tiles (J < I) have every j < i: mask word is zero.
    {
        const int wave = tid >> 5;
        const int lane = tid & 31;
        const int l16  = lane & 15;
        const bool lo  = (lane < 16);

        for (int tile = wave; tile < NTILE * NTILE; tile += 32) {
            int I = tile >> 6;
            int J = tile & 63;

            if (J < I) {                         // wave-uniform branch
                if (lane < 16)
                    mask16[(I * 16 + lane) * 64 + J] = 0;
                continue;
            }

            v2f A, B;
            A.x = lo ? areas[I * 16 + l16] : 0.0f;   // K=0 column: area_i
            A.y = lo ? 1.0f : 0.0f;                  // K=1 column: ones
            B.x = lo ? 1.0f : areas[J * 16 + l16];   // row K=0: ones, K=1: area_j
            B.y = 0.0f;                              // rows K=2,3: zeros
            v8f C = {0.f, 0.f, 0.f, 0.f, 0.f, 0.f, 0.f, 0.f};

            v8f S = __builtin_amdgcn_wmma_f32_16x16x4_f32(
                false, A, false, B, (short)0, C, false, false);

            int j = J * 16 + l16;
            const float4 bj = boxes4[j];             // ds_load_b128

            unsigned bal[8];
            int ibase = I * 16 + (lo ? 0 : 8);
#pragma unroll
            for (int r = 0; r < 8; ++r) {
                int i = ibase + r;
                const float4 bi = boxes4[i];         // ds_load_b128
                float xx1 = fmaxf(bi.x, bj.x), yy1 = fmaxf(bi.y, bj.y);
                float xx2 = fminf(bi.z, bj.z), yy2 = fminf(bi.w, bj.w);
                float iw = fmaxf(xx2 - xx1, 0.0f);
                float ih = fmaxf(yy2 - yy1, 0.0f);
                float inter = iw * ih;
                // iou > T  <=>  inter > T*(area_i+area_j-inter+eps); denom > 0
                bool cond = (inter > IOU_T * (S[r] - inter + 1e-7f)) && (j > i);
                bal[r] = __builtin_amdgcn_ballot_w32(cond);
            }
            if (lane == 0) {
#pragma unroll
                for (int r = 0; r < 8; ++r) {
                    // low 16 ballot bits = row I*16+r, high 16 = row I*16+r+8
                    mask16[(I * 16 + r) * 64 + J]     = (unsigned short)(bal[r] & 0xFFFFu);
                    mask16[(I * 16 + r + 8) * 64 + J] = (unsigned short)(bal[r] >> 16);
                }
            }
        }
    }
    __syncthreads();

    // ---------------- Phase E: serial suppression scan (one wave) ----------
    if (tid < 32) {
        unsigned kw = 0u;
        for (int c2 = 0; c2 < 32; ++c2)
            kw |= (scores[tid * 32 + c2] > 0.0f ? 1u : 0u) << c2;
        for (int i = 0; i < 1024; ++i) {
            unsigned wi = __shfl(kw, i >> 5, 32);       // broadcast keep word of i
            if ((wi >> (i & 31)) & 1u) {                 // uniform branch
                unsigned sup = (unsigned)mask16[i * 64 + 2 * tid] |
                               ((unsigned)mask16[i * 64 + 2 * tid + 1] << 16);
                kw &= ~sup;                              // mask has j>i baked in
            }
        }
        keepW[tid] = kw;
    }
    __syncthreads();

    // ---------------- Phase F: emit det * keep and keep flags ---------------
    {
        unsigned kb = (keepW[tid >> 5] >> (tid & 31)) & 1u;
        float4* dout4 = reinterpret_cast<float4*>(
            outDet + ((size_t)b * TOPK + tid) * 16);     // 64B-aligned row
        if (kb) {
            const float4* row4 =
                reinterpret_cast<const float4*>(P + (size_t)idxs[tid] * 16);
            float4 q1 = row4[1];    // elems 4..7  : obj, lmk0..2
            float4 q2 = row4[2];    // elems 8..11 : lmk3..6
            float4 q3 = row4[3];    // elems 12..15: lmk7..9, cls
            dout4[0] = boxes4[tid];
            dout4[1] = make_float4(scores[tid], q1.y, q1.z, q1.w);
            dout4[2] = q2;
            dout4[3] = make_float4(q3.x, q3.y, q3.z, 0.0f);
        } else {
            float4 z = make_float4(0.f, 0.f, 0.f, 0.f);
            dout4[0] = z; dout4[1] = z; dout4[2] = z; dout4[3] = z;
        }
        outKeep[(size_t)b * TOPK + tid] = kb ? 1.0f : 0.0f;
    }
}

extern "C" void kernel_launch(void* const* d_in, const int* in_sizes, int n_in,
                              void* d_out, int out_size, void* d_ws, size_t ws_size,
                              hipStream_t stream) {
    (void)in_sizes; (void)n_in; (void)d_ws; (void)ws_size; (void)out_size;
    const float* pred = (const float*)d_in[0];
    float* det  = (float*)d_out;                              // 32*1024*16 floats
    float* keep = (float*)d_out + (size_t)32 * TOPK * 16;     // 32*1024 flags

    // 256KB dynamic LDS per workgroup (CDNA5 allows up to 320KB/WGP).
    static const size_t kLds = 262144;
    (void)hipFuncSetAttribute((const void*)nms_gfx1250_kernel,
                              hipFuncAttributeMaxDynamicSharedMemorySize,
                              (int)kLds);
    nms_gfx1250_kernel<<<dim3(32), dim3(1024), kLds, stream>>>(pred, det, keep);
}